// LearnedSimulator_36344013258799
// MI455X (gfx1250) — compile-verified
//
#include <hip/hip_runtime.h>
#include <hip/hip_bf16.h>
#include <math.h>

// ---------------------------------------------------------------------------
// Problem constants (from reference)
// ---------------------------------------------------------------------------
#define NN        16384
#define TT        6
#define DIMS      3
#define EMB       16
#define MAXNB     128
#define RADIUS_F  0.08f
#define EPS_F     1e-8f
#define CAP       256          // per-row candidate capacity (expected ~35 hits)
#define WAVES_PB  4            // waves per block in the edge kernel

typedef __attribute__((ext_vector_type(2))) float v2f;
typedef __attribute__((ext_vector_type(8))) float v8f;

// ---------------------------------------------------------------------------
// Kernel A: node features + packed pos4 + squared norms
//   node_features[i] = [ (vel - mean)/std (15), emb[type[i]] (16) ]
// ---------------------------------------------------------------------------
__global__ __launch_bounds__(256)
void ls_nodes_kernel(const float* __restrict__ ps,      // (N, 6, 3)
                     const int*   __restrict__ ptype,   // (N,)
                     const float* __restrict__ emb,     // (9, 16)
                     const float* __restrict__ vmean,   // (3,)
                     const float* __restrict__ vstd,    // (3,)
                     float*       __restrict__ nf_out,  // (N, 31)
                     float4*      __restrict__ pos4,    // ws: (N,) xyz0
                     float*       __restrict__ sq)      // ws: (N,)
{
    int i = blockIdx.x * blockDim.x + threadIdx.x;
    if (i >= NN) return;

    const float* p = ps + (size_t)i * (TT * DIMS);
    float vm0 = vmean[0], vm1 = vmean[1], vm2 = vmean[2];
    float vs0 = vstd[0],  vs1 = vstd[1],  vs2 = vstd[2];

    float* nf = nf_out + (size_t)i * 31;
#pragma unroll
    for (int t = 0; t < TT - 1; ++t) {
        float vx = p[(t + 1) * 3 + 0] - p[t * 3 + 0];
        float vy = p[(t + 1) * 3 + 1] - p[t * 3 + 1];
        float vz = p[(t + 1) * 3 + 2] - p[t * 3 + 2];
        nf[t * 3 + 0] = (vx - vm0) / vs0;
        nf[t * 3 + 1] = (vy - vm1) / vs1;
        nf[t * 3 + 2] = (vz - vm2) / vs2;
    }
    int ty = ptype[i];
    const float* e = emb + (size_t)ty * EMB;
#pragma unroll
    for (int k = 0; k < EMB; ++k) nf[15 + k] = e[k];

    float x = p[15], y = p[16], z = p[17];      // position_sequence[:, -1]
    pos4[i] = make_float4(x, y, z, 0.0f);
    sq[i]   = x * x + y * y + z * z;
}

// ---------------------------------------------------------------------------
// Kernel B: radius search via V_WMMA_F32_16X16X4_F32 Gram tiles + LDS top-k.
// Block = 4 waves, all covering the SAME 16 receiver rows; each wave scans a
// disjoint quarter of the 16384 candidate columns and appends within-radius
// hits to shared per-row LDS lists (ds_add_rtn_u32). The final rank sort is
// order-independent, so cross-wave interleaving stays deterministic.
//
// WMMA f32 16x16x4 VGPR layout (ISA 7.12.2):
//   A (16x4): lanes 0-15 -> M=lane,   VGPR0=K0, VGPR1=K1
//             lanes16-31 -> M=lane-16,VGPR0=K2, VGPR1=K3
//   B (4x16): symmetric (row K striped across lanes within a VGPR)
//   C/D (16x16): VGPR v, lanes 0-15 -> M=v, lanes 16-31 -> M=v+8; N=lane%16
// ---------------------------------------------------------------------------
__global__ __launch_bounds__(WAVES_PB * 32)
void ls_edges_kernel(const float4* __restrict__ pos4,
                     const float*  __restrict__ sq,
                     float* __restrict__ senders_out,    // (N*128,) as float
                     float* __restrict__ receivers_out,  // (N*128,) as float
                     float* __restrict__ ef_out,         // (N*128, 4)
                     float* __restrict__ valid_out)      // (N*128,)
{
    __shared__ int   s_cnt[16];
    __shared__ float s_d2[16 * CAP];
    __shared__ int   s_j [16 * CAP];

    const int tid  = threadIdx.x;          // 0..127
    const int lane = tid & 31;
    const int wave = tid >> 5;             // 0..3
    const int half = lane >> 4;            // 0 | 1
    const int lmod = lane & 15;
    const int ic   = blockIdx.x * 16;      // receiver rows of this block

    const float thresh = (RADIUS_F + EPS_F) * (RADIUS_F + EPS_F);

    if (tid < 16) s_cnt[tid] = 0;
    __syncthreads();

    // Loop-invariant A tile, pre-scaled by -2 (so D = -2*P*P^T):
    float4 pa = pos4[ic + lmod];
    v2f a;
    a.x = -2.0f * (half ? pa.z : pa.x);
    a.y = half ? 0.0f : (-2.0f * pa.y);    // K3 pad = 0

    // sq of the receiver row held by accumulator VGPR v in this lane:
    float sqrow[8];
#pragma unroll
    for (int v = 0; v < 8; ++v) sqrow[v] = sq[ic + v + 8 * half];

    // This wave's column range (disjoint quarters of [0, NN)):
    const int jbeg = wave * (NN / WAVES_PB);
    const int jend = jbeg + (NN / WAVES_PB);

    // Software pipeline: B tile for iteration jc is loaded one step ahead.
    // The next-tile index is CLAMPED (not guarded) so the load is
    // unconditional every iteration -> no exec-mask round trip in the loop;
    // the final iteration re-loads the last (in-bounds) tile harmlessly.
    float4 pb    = pos4[jbeg + lmod];
    float  sqcol = sq[jbeg + lmod];

    for (int jc = jbeg; jc < jend; jc += 16) {
        __builtin_prefetch(&pos4[jc + 64], 0, 1);   // global_prefetch_b8

        // current tile operands
        v2f b;
        b.x = half ? pb.z : pb.x;
        b.y = half ? 0.0f : pb.y;
        float sqc = sqcol;

        // unconditionally load next tile into registers (clamped index)
        int jn = jc + 16;
        jn = (jn < jend) ? jn : (jend - 16);
        pb    = pos4[jn + lmod];
        sqcol = sq[jn + lmod];

        v8f c = {};
        // 8 args: (neg_a, A, neg_b, B, c_mod, C, reuse_a, reuse_b)
        c = __builtin_amdgcn_wmma_f32_16x16x4_f32(
                false, a, false, b, (short)0, c, false, false);

        const int j = jc + lmod;           // candidate sender (column N=lmod)
#pragma unroll
        for (int v = 0; v < 8; ++v) {
            float d2 = c[v] + sqrow[v] + sqc;
            if (d2 <= thresh) {
                int rl  = v + 8 * half;    // local receiver row 0..15
                int idx = atomicAdd(&s_cnt[rl], 1);   // ds_add_rtn_u32
                if (idx < CAP) {
                    s_d2[rl * CAP + idx] = d2;
                    s_j [rl * CAP + idx] = j;
                }
            }
        }
    }
    __syncthreads();

    // Phase 2 (all 128 threads): per-row rank sort
    // (d2 ascending, tie -> lower index; matches jax.lax.top_k semantics)
    for (int r = 0; r < 16; ++r) {
        const int row = ic + r;
        int cnt = s_cnt[r];
        if (cnt > CAP) cnt = CAP;
        const int nvalid = cnt < MAXNB ? cnt : MAXNB;
        const float4 pr = pos4[row];

        for (int m = tid; m < cnt; m += WAVES_PB * 32) {
            const float dm = s_d2[r * CAP + m];
            const int   jm = s_j [r * CAP + m];
            int rank = 0;
            for (int t = 0; t < cnt; ++t) {
                float dt = s_d2[r * CAP + t];
                int   jt = s_j [r * CAP + t];
                rank += (dt < dm) || (dt == dm && jt < jm);
            }
            if (rank < MAXNB) {
                const int e = row * MAXNB + rank;
                float4 psn = pos4[jm];
                float dx = (psn.x - pr.x) * (1.0f / RADIUS_F);
                float dy = (psn.y - pr.y) * (1.0f / RADIUS_F);
                float dz = (psn.z - pr.z) * (1.0f / RADIUS_F);
                float dist = sqrtf(dx * dx + dy * dy + dz * dz);
                senders_out[e]    = (float)jm;
                receivers_out[e]  = (float)row;
                ef_out[4 * e + 0] = dx;
                ef_out[4 * e + 1] = dy;
                ef_out[4 * e + 2] = dz;
                ef_out[4 * e + 3] = dist;
                valid_out[e]      = 1.0f;
            }
        }
        // pad remaining slots: sender=receiver, zero features, invalid
        for (int k = nvalid + tid; k < MAXNB; k += WAVES_PB * 32) {
            const int e = row * MAXNB + k;
            senders_out[e]    = (float)row;
            receivers_out[e]  = (float)row;
            ef_out[4 * e + 0] = 0.0f;
            ef_out[4 * e + 1] = 0.0f;
            ef_out[4 * e + 2] = 0.0f;
            ef_out[4 * e + 3] = 0.0f;
            valid_out[e]      = 0.0f;
        }
    }
}

// ---------------------------------------------------------------------------
// Host-side launch
// ---------------------------------------------------------------------------
extern "C" void kernel_launch(void* const* d_in, const int* in_sizes, int n_in,
                              void* d_out, int out_size, void* d_ws, size_t ws_size,
                              hipStream_t stream)
{
    (void)in_sizes; (void)n_in; (void)out_size; (void)ws_size;

    const float* ps    = (const float*)d_in[0];   // (N,6,3)
    const int*   ptype = (const int*)  d_in[1];   // (N,)
    const float* emb   = (const float*)d_in[2];   // (9,16)
    const float* vmean = (const float*)d_in[3];   // (3,)
    const float* vstd  = (const float*)d_in[4];   // (3,)

    float* out = (float*)d_out;
    // output layout: node_features | senders | receivers | edge_features | valid
    float* nf        = out;
    float* senders   = out + (size_t)NN * 31;
    float* receivers = senders + (size_t)NN * MAXNB;
    float* ef        = receivers + (size_t)NN * MAXNB;
    float* valid     = ef + (size_t)NN * MAXNB * 4;

    // workspace: pos4 (N float4) then sq (N float)
    float4* pos4 = (float4*)d_ws;
    float*  sqv  = (float*)(pos4 + NN);

    ls_nodes_kernel<<<(NN + 255) / 256, 256, 0, stream>>>(
        ps, ptype, emb, vmean, vstd, nf, pos4, sqv);

    ls_edges_kernel<<<NN / 16, WAVES_PB * 32, 0, stream>>>(
        pos4, sqv, senders, receivers, ef, valid);
}